// QuantumSimulator_3994319585571
// MI455X (gfx1250) — compile-verified
//
#include <hip/hip_runtime.h>
#include <math.h>

// CDNA5 WMMA operand types (wave32): 16x16x4 f32 -> A/B are 2 VGPRs, C/D are 8 VGPRs
typedef __attribute__((ext_vector_type(2))) float v2f;
typedef __attribute__((ext_vector_type(8))) float v8f;

#define NP  5
#define QDIM 32

// One block = 256 threads = 8 waves; each wave owns 16 batch rows.
// Phase 1: logits = pillars @ theta via two V_WMMA_F32_16X16X4_F32 (K=0..3, K=4 padded).
// Phase 2: analytic product-state amplitudes, closed-form 2x2 entanglement, softmax.
// Transcendentals use HW TRANS ops (v_sin/v_cos/v_exp/v_log): inputs are O(1), so
// accuracy is well within the float32 expm reference's own error.
__global__ __launch_bounds__(256) void qsim_fused_kernel(
    const float* __restrict__ pillars,   // (b, 5)
    const float* __restrict__ theta,     // (5, 5)
    float* __restrict__ out_states,      // (b, 32) complex64 -> 2*b*32 floats interleaved
    float* __restrict__ out_ent,         // (b,)
    float* __restrict__ out_act,         // (b, 5)
    int b)
{
    __shared__ float lds_logits[8][16][6];   // [wave][row][col], padded to 6

    const int t    = threadIdx.x;
    const int wave = t >> 5;
    const int lane = t & 31;
    const int block_row0 = blockIdx.x * 128;
    const int wave_row0  = block_row0 + wave * 16;

    // theta is tiny and reused by every wave: prefetch into cache hierarchy
    __builtin_prefetch(theta, 0, 0);

    // ---------------- Phase 1: WMMA GEMM (16 rows x 5 cols, K=5) ----------------
    // A 16x4 f32 layout: lanes 0-15 -> K={0,1} in vgpr{0,1}; lanes 16-31 -> K={2,3}
    const int m    = lane & 15;          // tile row
    const int half = lane >> 4;          // K-half selector
    const int kb   = half * 2;
    int r = wave_row0 + m;
    int rr = (r < b) ? r : (b - 1);      // clamp loads; stores guarded later

    v2f a1, a2, b1, b2;
    a1.x = pillars[rr * NP + kb + 0];
    a1.y = pillars[rr * NP + kb + 1];
    // second WMMA covers K'=4..7; only K'=4 (lanes 0-15, vgpr0) is real data
    a2.x = (half == 0) ? pillars[rr * NP + 4] : 0.0f;
    a2.y = 0.0f;

    // B 4x16 f32 (mirrored layout): col = lane&15, K = kb + vgpr
    const int n = lane & 15;
    b1.x = (n < NP) ? theta[(kb + 0) * NP + n] : 0.0f;
    b1.y = (n < NP) ? theta[(kb + 1) * NP + n] : 0.0f;
    b2.x = (half == 0 && n < NP) ? theta[4 * NP + n] : 0.0f;
    b2.y = 0.0f;

    v8f c = {};
    // 8 args: (neg_a, A, neg_b, B, c_mod, C, reuse_a, reuse_b)
    c = __builtin_amdgcn_wmma_f32_16x16x4_f32(false, a1, false, b1, (short)0, c, false, false);
    c = __builtin_amdgcn_wmma_f32_16x16x4_f32(false, a2, false, b2, (short)0, c, false, false);

    // C layout: vgpr v, lanes 0-15 -> M=v, N=lane; lanes 16-31 -> M=8+v, N=lane-16
    if (n < NP) {
        #pragma unroll
        for (int v = 0; v < 8; ++v) {
            lds_logits[wave][v + half * 8][n] = c[v];
        }
    }
    __syncthreads();

    // ---------------- Phase 2: states / entanglement / softmax ----------------
    // Two threads per row: hi=0 -> amplitudes 0..15 (+ scalars), hi=1 -> 16..31
    const int rl  = t >> 1;              // local row 0..127
    const int hi  = t & 1;
    const int row = block_row0 + rl;
    if (row < b) {
        float cs[NP], sn[NP];
        #pragma unroll
        for (int p = 0; p < NP; ++p) {
            float a = pillars[row * NP + p];
            sn[p] = __sinf(a);           // v_sin_f32 (TRANS)
            cs[p] = __cosf(a);           // v_cos_f32 (TRANS)
        }

        // product-state amplitudes: state[i] = prod_p (bit_p(i) ? sin : cos)
        // pack two complex64 values per 16B store -> global_store_b128
        float f4 = hi ? sn[4] : cs[4];   // bit 4 of i == hi
        float4* dst = (float4*)(out_states + (size_t)row * 2 * QDIM + (size_t)hi * QDIM);
        #pragma unroll
        for (int k = 0; k < 16; k += 2) {
            float v0 = f4, v1 = f4;
            v0 *= cs[0];                 v1 *= sn[0];           // k even / k+1 odd: bit0
            float b1f = (k & 2) ? sn[1] : cs[1];
            float b2f = (k & 4) ? sn[2] : cs[2];
            float b3f = (k & 8) ? sn[3] : cs[3];
            float rest = b1f * b2f * b3f;
            v0 *= rest; v1 *= rest;
            float4 z; z.x = v0; z.y = 0.0f; z.z = v1; z.w = 0.0f;
            dst[k >> 1] = z;
        }

        if (hi == 0) {
            // rho_A of a product state: rank-1 2x2, tr-normalized, eigvals closed-form
            float W = 1.0f;
            #pragma unroll
            for (int p = 0; p < 4; ++p) W *= (cs[p] * cs[p] + sn[p] * sn[p]);
            float r00 = cs[4] * cs[4] * W;
            float r11 = sn[4] * sn[4] * W;
            float r01 = cs[4] * sn[4] * W;
            float tr  = r00 + r11;
            float ent;
            if (tr > 0.0f) {
                float A  = r00 / tr, Dg = r11 / tr, O = r01 / tr;
                float mm  = 0.5f * (A + Dg);
                float det = A * Dg - O * O;
                float disc = __fsqrt_rn(fmaxf(mm * mm - det, 0.0f));
                float e0 = mm - disc, e1 = mm + disc;
                ent = 0.0f;
                if (e0 > 1e-9f) ent -= e0 * __logf(e0);
                if (e1 > 1e-9f) ent -= e1 * __logf(e1);
            } else {
                ent = 0.6931471805599453f;           // rho = I/2 -> log 2
            }
            out_ent[row] = ent;

            // softmax over the 5 WMMA logits
            float l[NP];
            #pragma unroll
            for (int j = 0; j < NP; ++j) l[j] = lds_logits[rl >> 4][rl & 15][j];
            float mx = l[0];
            #pragma unroll
            for (int j = 1; j < NP; ++j) mx = fmaxf(mx, l[j]);
            float e[NP], s = 0.0f;
            #pragma unroll
            for (int j = 0; j < NP; ++j) { e[j] = __expf(l[j] - mx); s += e[j]; }
            float inv = 1.0f / s;
            #pragma unroll
            for (int j = 0; j < NP; ++j) out_act[(size_t)row * NP + j] = e[j] * inv;
        }
    }
}

extern "C" void kernel_launch(void* const* d_in, const int* in_sizes, int n_in,
                              void* d_out, int out_size, void* d_ws, size_t ws_size,
                              hipStream_t stream) {
    (void)n_in; (void)out_size; (void)d_ws; (void)ws_size;
    const float* pillars = (const float*)d_in[0];   // (b, 5) float32
    const float* theta   = (const float*)d_in[1];   // (5, 5) float32
    const int b = in_sizes[0] / NP;

    float* out        = (float*)d_out;
    float* out_states = out;                         // 2*b*32 floats (complex64)
    float* out_ent    = out + (size_t)b * 2 * QDIM;  // b floats
    float* out_act    = out_ent + b;                 // b*5 floats

    const int blocks = (b + 127) / 128;              // 128 rows per 256-thread block
    qsim_fused_kernel<<<blocks, 256, 0, stream>>>(pillars, theta,
                                                  out_states, out_ent, out_act, b);
}